// H2GCN_55722905699171
// MI455X (gfx1250) — compile-verified
//
#include <hip/hip_runtime.h>
#include <hip/hip_bf16.h>

typedef __attribute__((ext_vector_type(2))) float v2f;
typedef __attribute__((ext_vector_type(8))) float v8f;

#define WMMA_F32X4(A, B, C) \
    __builtin_amdgcn_wmma_f32_16x16x4_f32(false, (A), false, (B), (short)0, (C), false, false)

// ---------------------------------------------------------------------------
// h0 = relu(x @ W_embed + b_embed)   x:[n,512] W:[512,64] -> h0:[n,64]
// One wave computes a 16-row x 64-col strip via 4 WMMA accumulators, so x is
// streamed exactly once; W_embed (128KB) stays L2-resident.
// ---------------------------------------------------------------------------
__global__ __launch_bounds__(256)
void h2gcn_embed_wmma(const float* __restrict__ x, const float* __restrict__ W,
                      const float* __restrict__ bias, float* __restrict__ h0, int n) {
    const int lane = threadIdx.x & 31;
    const int wave = threadIdx.x >> 5;
    const int half = lane >> 4;    // 0: K pair {0,1}, 1: K pair {2,3}
    const int mrow = lane & 15;    // A row / B,C,D column-in-tile
    const int row0 = (blockIdx.x * 8 + wave) * 16;

    int rowc = row0 + mrow;
    if (rowc >= n) rowc = n - 1;                   // clamp loads, guard stores
    const float* xrow = x + (size_t)rowc * 512;

    v8f acc0 = {}, acc1 = {}, acc2 = {}, acc3 = {};
    for (int k = 0; k < 512; k += 4) {
        const int k0 = k + 2 * half;
        v2f a = *(const v2f*)(xrow + k0);
        v2f b0, b1, b2, b3;
        b0.x = W[k0 * 64 +  0 + mrow]; b0.y = W[(k0 + 1) * 64 +  0 + mrow];
        b1.x = W[k0 * 64 + 16 + mrow]; b1.y = W[(k0 + 1) * 64 + 16 + mrow];
        b2.x = W[k0 * 64 + 32 + mrow]; b2.y = W[(k0 + 1) * 64 + 32 + mrow];
        b3.x = W[k0 * 64 + 48 + mrow]; b3.y = W[(k0 + 1) * 64 + 48 + mrow];
        acc0 = WMMA_F32X4(a, b0, acc0);
        acc1 = WMMA_F32X4(a, b1, acc1);
        acc2 = WMMA_F32X4(a, b2, acc2);
        acc3 = WMMA_F32X4(a, b3, acc3);
    }

    const float bi0 = bias[ 0 + mrow], bi1 = bias[16 + mrow];
    const float bi2 = bias[32 + mrow], bi3 = bias[48 + mrow];
#pragma unroll
    for (int v = 0; v < 8; ++v) {
        const int r = row0 + v + 8 * half;
        if (r < n) {
            float* o = h0 + (size_t)r * 64 + mrow;
            float t;
            t = acc0[v] + bi0; o[ 0] = t > 0.f ? t : 0.f;
            t = acc1[v] + bi1; o[16] = t > 0.f ? t : 0.f;
            t = acc2[v] + bi2; o[32] = t > 0.f ? t : 0.f;
            t = acc3[v] + bi3; o[48] = t > 0.f ? t : 0.f;
        }
    }
}

// ---------------------------------------------------------------------------
// SpMM: hout[dst, col_off + :D] += hin[src, :D]   (binary adjacency)
// One thread moves one float4 chunk: coalesced gather + 4 no-return fp32
// atomics. Source matrices fit in the 192MB L2, so this is L2-bandwidth work.
// ---------------------------------------------------------------------------
__global__ __launch_bounds__(256)
void h2gcn_spmm_atomic(const int* __restrict__ src, const int* __restrict__ dst,
                       const float* __restrict__ hin, float* __restrict__ hout,
                       int e, int log_chunks, int out_stride, int col_off) {
    const int idx = blockIdx.x * 256 + threadIdx.x;
    const int total = e << log_chunks;
    if (idx >= total) return;
    const int edge = idx >> log_chunks;
    const int c    = (idx & ((1 << log_chunks) - 1)) * 4;
    const int D    = 4 << log_chunks;
    const int s = src[edge], d = dst[edge];
    const float4 v = *(const float4*)(hin + (size_t)s * D + c);
    float* o = hout + (size_t)d * out_stride + col_off + c;
    (void)__hip_atomic_fetch_add(o + 0, v.x, __ATOMIC_RELAXED, __HIP_MEMORY_SCOPE_AGENT);
    (void)__hip_atomic_fetch_add(o + 1, v.y, __ATOMIC_RELAXED, __HIP_MEMORY_SCOPE_AGENT);
    (void)__hip_atomic_fetch_add(o + 2, v.z, __ATOMIC_RELAXED, __HIP_MEMORY_SCOPE_AGENT);
    (void)__hip_atomic_fetch_add(o + 3, v.w, __ATOMIC_RELAXED, __HIP_MEMORY_SCOPE_AGENT);
}

// ---------------------------------------------------------------------------
// logits = concat(h0,h1,h2) @ W_cls + b_cls  -> d_out [n,40]
// K = 64 + 128 + 256 = 448, three column tiles (48 cols, last 8 masked).
// ---------------------------------------------------------------------------
__global__ __launch_bounds__(256)
void h2gcn_cls_wmma(const float* __restrict__ h0, const float* __restrict__ h1,
                    const float* __restrict__ h2, const float* __restrict__ W,
                    const float* __restrict__ bias, float* __restrict__ logits, int n) {
    const int lane = threadIdx.x & 31;
    const int wave = threadIdx.x >> 5;
    const int half = lane >> 4;
    const int mrow = lane & 15;
    const int row0 = (blockIdx.x * 8 + wave) * 16;

    int rowc = row0 + mrow;
    if (rowc >= n) rowc = n - 1;
    const float* a0 = h0 + (size_t)rowc * 64;
    const float* a1 = h1 + (size_t)rowc * 128;
    const float* a2 = h2 + (size_t)rowc * 256;

    const int n0 = mrow, n1 = 16 + mrow, n2 = 32 + mrow;
    const bool n2v = (n2 < 40);

    v8f acc0 = {}, acc1 = {}, acc2 = {};

#define CLS_KSTEP(APTR, KW)                                                       \
    {                                                                             \
        const int k0 = KW + 2 * half;                                             \
        v2f a = *(const v2f*)((APTR) + k0);                                       \
        v2f b0, b1, b2;                                                           \
        b0.x = W[k0 * 40 + n0]; b0.y = W[(k0 + 1) * 40 + n0];                     \
        b1.x = W[k0 * 40 + n1]; b1.y = W[(k0 + 1) * 40 + n1];                     \
        b2.x = n2v ? W[k0 * 40 + n2] : 0.f;                                       \
        b2.y = n2v ? W[(k0 + 1) * 40 + n2] : 0.f;                                 \
        acc0 = WMMA_F32X4(a, b0, acc0);                                           \
        acc1 = WMMA_F32X4(a, b1, acc1);                                           \
        acc2 = WMMA_F32X4(a, b2, acc2);                                           \
    }

    for (int k = 0; k < 64; k += 4)  CLS_KSTEP(a0, k);       // W rows [0,64)
    for (int k = 0; k < 128; k += 4) { const float* Wp = W + 64 * 40;
        const int kw = k; { const int k0 = kw + 2 * half;
            v2f a = *(const v2f*)(a1 + k0);
            v2f b0, b1, b2;
            b0.x = Wp[k0 * 40 + n0]; b0.y = Wp[(k0 + 1) * 40 + n0];
            b1.x = Wp[k0 * 40 + n1]; b1.y = Wp[(k0 + 1) * 40 + n1];
            b2.x = n2v ? Wp[k0 * 40 + n2] : 0.f;
            b2.y = n2v ? Wp[(k0 + 1) * 40 + n2] : 0.f;
            acc0 = WMMA_F32X4(a, b0, acc0);
            acc1 = WMMA_F32X4(a, b1, acc1);
            acc2 = WMMA_F32X4(a, b2, acc2);
        } }
    for (int k = 0; k < 256; k += 4) { const float* Wp = W + 192 * 40;
        const int kw = k; { const int k0 = kw + 2 * half;
            v2f a = *(const v2f*)(a2 + k0);
            v2f b0, b1, b2;
            b0.x = Wp[k0 * 40 + n0]; b0.y = Wp[(k0 + 1) * 40 + n0];
            b1.x = Wp[k0 * 40 + n1]; b1.y = Wp[(k0 + 1) * 40 + n1];
            b2.x = n2v ? Wp[k0 * 40 + n2] : 0.f;
            b2.y = n2v ? Wp[(k0 + 1) * 40 + n2] : 0.f;
            acc0 = WMMA_F32X4(a, b0, acc0);
            acc1 = WMMA_F32X4(a, b1, acc1);
            acc2 = WMMA_F32X4(a, b2, acc2);
        } }
#undef CLS_KSTEP

    const float bi0 = bias[n0], bi1 = bias[n1], bi2 = n2v ? bias[n2] : 0.f;
#pragma unroll
    for (int v = 0; v < 8; ++v) {
        const int r = row0 + v + 8 * half;
        if (r < n) {
            float* o = logits + (size_t)r * 40;
            o[n0] = acc0[v] + bi0;
            o[n1] = acc1[v] + bi1;
            if (n2v) o[n2] = acc2[v] + bi2;
        }
    }
}

// ---------------------------------------------------------------------------
// In-place row-wise log_softmax on [n,40]
// ---------------------------------------------------------------------------
__global__ __launch_bounds__(256)
void h2gcn_logsoftmax(float* __restrict__ out, int n) {
    const int i = blockIdx.x * 256 + threadIdx.x;
    if (i >= n) return;
    float* p = out + (size_t)i * 40;
    float m = -1e30f;
#pragma unroll
    for (int j = 0; j < 40; ++j) m = fmaxf(m, p[j]);
    float s = 0.f;
#pragma unroll
    for (int j = 0; j < 40; ++j) s += __expf(p[j] - m);
    const float ls = __logf(s);
#pragma unroll
    for (int j = 0; j < 40; ++j) p[j] = p[j] - m - ls;
}

__global__ __launch_bounds__(256)
void h2gcn_zero_f4(float4* __restrict__ p, long n4) {
    const long i = (long)blockIdx.x * 256 + threadIdx.x;
    if (i < n4) p[i] = make_float4(0.f, 0.f, 0.f, 0.f);
}

extern "C" void kernel_launch(void* const* d_in, const int* in_sizes, int n_in,
                              void* d_out, int out_size, void* d_ws, size_t ws_size,
                              hipStream_t stream) {
    const float* x       = (const float*)d_in[0];
    const float* W_embed = (const float*)d_in[1];
    const float* b_embed = (const float*)d_in[2];
    const float* W_cls   = (const float*)d_in[3];
    const float* b_cls   = (const float*)d_in[4];
    const int*   a1s     = (const int*)d_in[5];
    const int*   a1d     = (const int*)d_in[6];
    const int*   a2s     = (const int*)d_in[7];
    const int*   a2d     = (const int*)d_in[8];

    const int n = in_sizes[0] / 512;   // 100000
    const int e = in_sizes[5];         // 1600000

    float* h0 = (float*)d_ws;                    // [n,64]
    float* h1 = h0 + (size_t)n * 64;             // [n,128]
    float* h2 = h1 + (size_t)n * 128;            // [n,256]
    float* out = (float*)d_out;                  // [n,40]

    // Zero the SpMM accumulators (h1, h2 are contiguous: n*384 floats).
    const long z4 = (long)n * 384 / 4;
    h2gcn_zero_f4<<<(int)((z4 + 255) / 256), 256, 0, stream>>>((float4*)h1, z4);

    const int row_blocks = (n + 127) / 128;      // 8 waves x 16 rows per block
    h2gcn_embed_wmma<<<row_blocks, 256, 0, stream>>>(x, W_embed, b_embed, h0, n);

    // Round 1: h0 (D=64, 16 chunks/edge) -> h1 columns [0,64) and [64,128)
    const int t1 = e * 16;
    h2gcn_spmm_atomic<<<(t1 + 255) / 256, 256, 0, stream>>>(a1s, a1d, h0, h1, e, 4, 128, 0);
    h2gcn_spmm_atomic<<<(t1 + 255) / 256, 256, 0, stream>>>(a2s, a2d, h0, h1, e, 4, 128, 64);

    // Round 2: h1 (D=128, 32 chunks/edge) -> h2 columns [0,128) and [128,256)
    const int t2 = e * 32;
    h2gcn_spmm_atomic<<<(t2 + 255) / 256, 256, 0, stream>>>(a1s, a1d, h1, h2, e, 5, 256, 0);
    h2gcn_spmm_atomic<<<(t2 + 255) / 256, 256, 0, stream>>>(a2s, a2d, h1, h2, e, 5, 256, 128);

    h2gcn_cls_wmma<<<row_blocks, 256, 0, stream>>>(h0, h1, h2, W_cls, b_cls, out, n);
    h2gcn_logsoftmax<<<(n + 255) / 256, 256, 0, stream>>>(out, n);
}